// ConjGradLoss_ANorm_82978768159396
// MI455X (gfx1250) — compile-verified
//
#include <hip/hip_runtime.h>

typedef __attribute__((ext_vector_type(2))) float v2f;
typedef __attribute__((ext_vector_type(4))) float v4f;
typedef __attribute__((ext_vector_type(8))) float v8f;
typedef __attribute__((ext_vector_type(4))) int   v4i;

#define NGRAPH 64
#define EPSF   1e-6f

// ---------------------------------------------------------------------------
// Full-wave32 sum via V_WMMA_F32_16X16X4_F32.
// A: VGPR0=v, VGPR1=0  => A[m][0]=val[m], A[m][2]=val[m+16], others 0.
// B: all ones          => D[m][n] = val[m] + val[m+16]  (any n).
// Lane L holds D rows M = (L<16 ? 0..7 : 8..15) in its 8 VGPRs; summing those
// and adding the opposite half (shfl_xor 16) gives the 32-lane total on every
// lane. Requires EXEC == all ones (callers guarantee full waves).
// ---------------------------------------------------------------------------
__device__ __forceinline__ float wave_sum32_wmma(float v) {
  v2f a; a.x = v;    a.y = 0.0f;
  v2f b; b.x = 1.0f; b.y = 1.0f;
  v8f c = {};
  c = __builtin_amdgcn_wmma_f32_16x16x4_f32(false, a, false, b, (short)0, c,
                                            false, false);
  float s = c[0] + c[1] + c[2] + c[3] + c[4] + c[5] + c[6] + c[7];
  s += __shfl_xor(s, 16, 32);
  return s;
}

// ---------------------------------------------------------------------------
// Kernel 1: COO SpMV  Ad[row] += A*mask*d[col]
// Edge arrays are touched exactly once -> non-temporal B128 loads so the
// resident d / Ad vectors stay in the 192MB L2. Scatter via HW f32 atomics.
// ---------------------------------------------------------------------------
__global__ void k_spmv(const float* __restrict__ A, const float* __restrict__ M,
                       const int* __restrict__ rows, const int* __restrict__ cols,
                       const float* __restrict__ d, float* __restrict__ Ad,
                       int n_edges) {
  const int tid = blockIdx.x * blockDim.x + threadIdx.x;
  const int T   = gridDim.x * blockDim.x;
  const int n4  = n_edges >> 2;
  for (int i = tid; i < n4; i += T) {
    v4f a = __builtin_nontemporal_load((const v4f*)A + i);
    v4f m = __builtin_nontemporal_load((const v4f*)M + i);
    v4i r = __builtin_nontemporal_load((const v4i*)rows + i);
    v4i c = __builtin_nontemporal_load((const v4i*)cols + i);
    unsafeAtomicAdd(&Ad[r.x], a.x * m.x * d[c.x]);
    unsafeAtomicAdd(&Ad[r.y], a.y * m.y * d[c.y]);
    unsafeAtomicAdd(&Ad[r.z], a.z * m.z * d[c.z]);
    unsafeAtomicAdd(&Ad[r.w], a.w * m.w * d[c.w]);
  }
  // tail (n_edges not multiple of 4) - not hit for 2^24 edges
  for (int i = (n4 << 2) + tid; i < n_edges; i += T) {
    unsafeAtomicAdd(&Ad[rows[i]], A[i] * M[i] * d[cols[i]]);
  }
}

// ---------------------------------------------------------------------------
// Kernel 2: per-graph  sum(r*d), sum(d*Ad).  batch is sorted, so nearly every
// wave is graph-uniform: WMMA wave-reduce + 1 atomic/wave instead of 32.
// n_main is a multiple of total threads -> main loop always runs full waves.
// ---------------------------------------------------------------------------
__global__ void k_dots(const float* __restrict__ r, const float* __restrict__ d,
                       const float* __restrict__ Ad, const int* __restrict__ batch,
                       float* __restrict__ sums, int n, int n_main) {
  const int tid = blockIdx.x * blockDim.x + threadIdx.x;
  const int T   = gridDim.x * blockDim.x;
  for (int i = tid; i < n_main; i += T) {
    const float p0 = r[i] * d[i];
    const float p1 = d[i] * Ad[i];
    const int   g  = batch[i];
    const int   g0 = __builtin_amdgcn_readfirstlane(g);
    if (__all(g == g0)) {
      const float s0 = wave_sum32_wmma(p0);
      const float s1 = wave_sum32_wmma(p1);
      if ((threadIdx.x & 31) == 0) {
        unsafeAtomicAdd(&sums[g0], s0);
        unsafeAtomicAdd(&sums[NGRAPH + g0], s1);
      }
    } else {
      unsafeAtomicAdd(&sums[g], p0);
      unsafeAtomicAdd(&sums[NGRAPH + g], p1);
    }
  }
  for (int i = n_main + tid; i < n; i += T) {
    const int g = batch[i];
    unsafeAtomicAdd(&sums[g], r[i] * d[i]);
    unsafeAtomicAdd(&sums[NGRAPH + g], d[i] * Ad[i]);
  }
}

// ---------------------------------------------------------------------------
// Kernel 3: alpha[g] = r.d / (d.Ad + eps)
// ---------------------------------------------------------------------------
__global__ void k_alpha(const float* __restrict__ sums, float* __restrict__ alpha) {
  const int g = threadIdx.x;
  if (g < NGRAPH) alpha[g] = sums[g] / (sums[NGRAPH + g] + EPSF);
}

// ---------------------------------------------------------------------------
// Kernel 4: err = alpha[g]*Ad - r ; per-graph sum(err^2), sum(r^2)
// ---------------------------------------------------------------------------
__global__ void k_err(const float* __restrict__ r, const float* __restrict__ Ad,
                      const int* __restrict__ batch, const float* __restrict__ alpha,
                      float* __restrict__ sums, int n, int n_main) {
  const int tid = blockIdx.x * blockDim.x + threadIdx.x;
  const int T   = gridDim.x * blockDim.x;
  for (int i = tid; i < n_main; i += T) {
    const int   g  = batch[i];
    const float rv = r[i];
    const float e  = alpha[g] * Ad[i] - rv;
    const float p2 = e * e;
    const float p3 = rv * rv;
    const int   g0 = __builtin_amdgcn_readfirstlane(g);
    if (__all(g == g0)) {
      const float s2 = wave_sum32_wmma(p2);
      const float s3 = wave_sum32_wmma(p3);
      if ((threadIdx.x & 31) == 0) {
        unsafeAtomicAdd(&sums[2 * NGRAPH + g0], s2);
        unsafeAtomicAdd(&sums[3 * NGRAPH + g0], s3);
      }
    } else {
      unsafeAtomicAdd(&sums[2 * NGRAPH + g], p2);
      unsafeAtomicAdd(&sums[3 * NGRAPH + g], p3);
    }
  }
  for (int i = n_main + tid; i < n; i += T) {
    const int   g  = batch[i];
    const float rv = r[i];
    const float e  = alpha[g] * Ad[i] - rv;
    unsafeAtomicAdd(&sums[2 * NGRAPH + g], e * e);
    unsafeAtomicAdd(&sums[3 * NGRAPH + g], rv * rv);
  }
}

// ---------------------------------------------------------------------------
// Kernel 5: loss = mean_g( err_sq / (gt_sq + eps) )
// ---------------------------------------------------------------------------
__global__ void k_final(const float* __restrict__ sums, float* __restrict__ out) {
  __shared__ float sm[NGRAPH];
  const int g = threadIdx.x;
  if (g < NGRAPH)
    sm[g] = sums[2 * NGRAPH + g] / (sums[3 * NGRAPH + g] + EPSF);
  __syncthreads();
  if (g == 0) {
    float t = 0.0f;
    for (int j = 0; j < NGRAPH; ++j) t += sm[j];
    out[0] = t / (float)NGRAPH;
  }
}

// ---------------------------------------------------------------------------
// Host entry. Input order: d, residual, matrix_values, mask, L_values(unused),
// edge_index (2 x E, rows then cols), batch_vec. Output: 1 float.
// Workspace layout (floats): [0, n_nodes) = Ad ; then 4*64 sums ; then 64 alpha.
// ---------------------------------------------------------------------------
extern "C" void kernel_launch(void* const* d_in, const int* in_sizes, int n_in,
                              void* d_out, int out_size, void* d_ws, size_t ws_size,
                              hipStream_t stream) {
  const float* dvec  = (const float*)d_in[0];
  const float* resid = (const float*)d_in[1];
  const float* Avals = (const float*)d_in[2];
  const float* mask  = (const float*)d_in[3];
  const int*   eidx  = (const int*)d_in[5];
  const int*   batch = (const int*)d_in[6];

  const int n_nodes = in_sizes[0];
  const int n_edges = in_sizes[2];
  const int* rows = eidx;
  const int* cols = eidx + n_edges;

  float* ws    = (float*)d_ws;
  float* Ad    = ws;
  float* sums  = ws + n_nodes;            // 4 * 64
  float* alpha = sums + 4 * NGRAPH;       // 64

  hipMemsetAsync(d_ws, 0, (size_t)(n_nodes + 5 * NGRAPH) * sizeof(float), stream);

  const dim3 blk(256);
  k_spmv<<<2048, blk, 0, stream>>>(Avals, mask, rows, cols, dvec, Ad, n_edges);

  const int nblk2  = 512;
  const int T2     = nblk2 * 256;
  const int n_main = n_nodes - (n_nodes % T2);
  k_dots<<<nblk2, blk, 0, stream>>>(resid, dvec, Ad, batch, sums, n_nodes, n_main);
  k_alpha<<<1, 64, 0, stream>>>(sums, alpha);
  k_err<<<nblk2, blk, 0, stream>>>(resid, Ad, batch, alpha, sums, n_nodes, n_main);
  k_final<<<1, 64, 0, stream>>>(sums, (float*)d_out);
}